// VCDN_29377576304736
// MI455X (gfx1250) — compile-verified
//
#include <hip/hip_runtime.h>

// VCDN cross-feature classifier head for MI455X (gfx1250).
// out[c] = relu(b[c] + sum_d W[c,d] * t01[d/2304] * t23[d%2304])
// Bandwidth-bound: 340 MB of W streamed once (~14.6 us floor @ 23.3 TB/s).
// WMMA f32 16x16x4 does the 16-class GEMV accumulation; W is streamed with
// per-lane global_load_b128 (512 B per wave per load).

#define VIEW 48
#define T2   2304                 // 48*48
#define NC   16
#define DTOT (2304u * 2304u)      // 5,308,416
#define MAIN_BLOCKS      144
#define WAVES_PER_BLOCK  8
#define NWAVES (MAIN_BLOCKS * WAVES_PER_BLOCK)   // 1152 -> each wave gets 2 ij segments

typedef float v2f __attribute__((ext_vector_type(2)));
typedef float v4f __attribute__((ext_vector_type(4)));
typedef float v8f __attribute__((ext_vector_type(8)));

// ---------------------------------------------------------------------------
// Kernel 1: factor vectors t01 = outer(x0,x1).ravel(), t23 = outer(x2,x3).ravel()
// ---------------------------------------------------------------------------
__global__ __launch_bounds__(256) void vcdn_prep(const float* __restrict__ x,
                                                 float* __restrict__ t01,
                                                 float* __restrict__ t23) {
  int i = blockIdx.x * 256 + threadIdx.x;
  if (i < T2) {
    int r = i / VIEW;
    int c = i % VIEW;
    t01[i] = x[0 * VIEW + r] * x[1 * VIEW + c];
    t23[i] = x[2 * VIEW + r] * x[3 * VIEW + c];
  }
}

// ---------------------------------------------------------------------------
// Kernel 2: stream W, accumulate 16-class dot products with WMMA f32 16x16x4.
//
// K-slot remap for b128 streaming: for an 8-wide d chunk [d0, d0+8),
//   lanes 0-15  (hi=0) load W[m, d0   .. d0+3]  (one b128)
//   lanes 16-31 (hi=1) load W[m, d0+4 .. d0+7]  (one b128)
// wmma0 consumes (.x,.y): slots K0,K1 = d0,d0+1 (low half) and
//                         slots K2,K3 = d0+4,d0+5 (high half)
// wmma1 consumes (.z,.w): d0+2,d0+3 / d0+6,d0+7.
// B holds the matching cfdt values (same per-half indices, broadcast over N),
// so each d in [d0, d0+8) is paired with its own cf exactly once.
// ---------------------------------------------------------------------------
__global__ __launch_bounds__(256) void vcdn_main(const float* __restrict__ W,
                                                 const float* __restrict__ t01,
                                                 const float* __restrict__ t23g,
                                                 float* __restrict__ partial) {
  __shared__ float t23[T2];
  for (int i = threadIdx.x; i < T2; i += 256) t23[i] = t23g[i];
  __syncthreads();

  const int lane = threadIdx.x & 31;
  const int m    = lane & 15;        // class row this lane feeds into A
  const int hi   = lane >> 4;        // 0: low K half, 1: high K half
  const int wid  = blockIdx.x * WAVES_PER_BLOCK + (threadIdx.x >> 5);

  v8f acc0 = {0.f, 0.f, 0.f, 0.f, 0.f, 0.f, 0.f, 0.f};
  v8f acc1 = {0.f, 0.f, 0.f, 0.f, 0.f, 0.f, 0.f, 0.f};

  for (int ij = wid; ij < T2; ij += NWAVES) {
    const float s = t01[ij];                                  // per-segment scale
    const float* wp = W + (size_t)m * DTOT + (size_t)ij * T2 + 4 * hi;

    for (int klo = 0; klo < T2; klo += 64) {
      __builtin_prefetch(wp + klo + 1024, 0, 0);              // ~4 KB ahead per row stream
      #pragma unroll
      for (int kq = 0; kq < 64; kq += 8) {
        const int k0 = klo + kq;
        // A: 4 contiguous K-values for this lane's half (16B-aligned b128)
        v4f a = *(const v4f*)(wp + k0);
        // matching cfdt values (16B-aligned LDS b128, broadcast read)
        v4f t = *(const v4f*)&t23[k0 + 4 * hi];
        v2f b0; b0.x = s * t.x; b0.y = s * t.y;
        v2f b1; b1.x = s * t.z; b1.y = s * t.w;
        v2f a0; a0.x = a.x; a0.y = a.y;
        v2f a1; a1.x = a.z; a1.y = a.w;
        acc0 = __builtin_amdgcn_wmma_f32_16x16x4_f32(false, a0, false, b0,
                                                     (short)0, acc0, false, false);
        acc1 = __builtin_amdgcn_wmma_f32_16x16x4_f32(false, a1, false, b1,
                                                     (short)0, acc1, false, false);
      }
    }
  }

  // Combine the two accumulator chains; every N column holds the same dots.
  v8f acc;
  #pragma unroll
  for (int r = 0; r < 8; ++r) acc[r] = acc0[r] + acc1[r];

  // C/D layout: VGPR r, lane 0  -> (M=r,   N=0)
  //             VGPR r, lane 16 -> (M=8+r, N=0)
  float* p = partial + (size_t)wid * NC;
  if (lane == 0) {
    #pragma unroll
    for (int r = 0; r < 8; ++r) p[r] = acc[r];
  } else if (lane == 16) {
    #pragma unroll
    for (int r = 0; r < 8; ++r) p[8 + r] = acc[r];
  }
}

// ---------------------------------------------------------------------------
// Kernel 3: deterministic reduction of per-wave partials + bias + ReLU.
// ---------------------------------------------------------------------------
__global__ __launch_bounds__(256) void vcdn_reduce(const float* __restrict__ partial,
                                                   const float* __restrict__ bias,
                                                   float* __restrict__ out) {
  __shared__ float red[256];
  const int c   = threadIdx.x & 15;   // class
  const int sub = threadIdx.x >> 4;   // 16 sub-reducers per class
  float s = 0.f;
  for (int w = sub; w < NWAVES; w += 16) s += partial[(size_t)w * NC + c];
  red[threadIdx.x] = s;
  __syncthreads();
  if (threadIdx.x < NC) {
    float t = 0.f;
    #pragma unroll
    for (int k = 0; k < 16; ++k) t += red[k * NC + threadIdx.x];
    t += bias[threadIdx.x];
    out[threadIdx.x] = t > 0.f ? t : 0.f;
  }
}

// ---------------------------------------------------------------------------
extern "C" void kernel_launch(void* const* d_in, const int* in_sizes, int n_in,
                              void* d_out, int out_size, void* d_ws, size_t ws_size,
                              hipStream_t stream) {
  (void)in_sizes; (void)n_in; (void)out_size; (void)ws_size;
  const float* x = (const float*)d_in[0];   // (4, 48)
  const float* W = (const float*)d_in[1];   // (16, D) row-major
  const float* b = (const float*)d_in[2];   // (16,)
  float* out = (float*)d_out;               // (16,)

  float* t01     = (float*)d_ws;            // 2304 floats
  float* t23     = t01 + T2;                // 2304 floats
  float* partial = t23 + T2;                // NWAVES*16 floats (73,728 B)

  vcdn_prep  <<<(T2 + 255) / 256, 256, 0, stream>>>(x, t01, t23);
  vcdn_main  <<<MAIN_BLOCKS,      256, 0, stream>>>(W, t01, t23, partial);
  vcdn_reduce<<<1,                256, 0, stream>>>(partial, b, out);
}